// SimpleEdgeNetRegression_5858335391873
// MI455X (gfx1250) — compile-verified
//
#include <hip/hip_runtime.h>
#include <hip/hip_bf16.h>
#include <math.h>

// ---------------------------------------------------------------------------
// SimpleEdgeNetRegression on MI455X (gfx1250, wave32)
//   x[N,4] -> edge MLP (8->32 relu ->32) via V_WMMA_F32_16X16X4_F32,
//   segment_max via native GLOBAL_ATOMIC_MAX_NUM_F32 (inline asm, exact,
//   order-independent => deterministic), batchnorm (deterministic two-pass),
//   range-based per-graph mean pool (batch is sorted), 32->16->1 head.
// ---------------------------------------------------------------------------

#define HID 32
#define LDS_STRIDE 34          // 16 x 34 floats per wave tile (bank-friendly)
#define WAVES_PER_BLOCK 8
#define EDGE_BLOCKS 1024
#define STAT_BLOCKS 256

typedef float v2f __attribute__((ext_vector_type(2)));
typedef float v8f __attribute__((ext_vector_type(8)));

__device__ __forceinline__ v8f wmma_f32_16x16x4(v2f a, v2f b, v8f c) {
  // 8 args: (neg_a, A, neg_b, B, c_mod, C, reuse_a, reuse_b)
  return __builtin_amdgcn_wmma_f32_16x16x4_f32(false, a, false, b,
                                               (short)0, c, false, false);
}

// Native CDNA5 float atomic max (no-return form, tracked on STOREcnt).
// IEEE maximumNumber: exact, commutative/associative => deterministic result.
// Destination must be initialized to -inf.
__device__ __forceinline__ void atomicMaxF(float* addr, float v) {
  asm volatile("global_atomic_max_num_f32 %0, %1, off"
               :
               : "v"(addr), "v"(v)
               : "memory");
}

// ---------------------------------------------------------------------------
// Kernel 0: fill agg with -inf
// ---------------------------------------------------------------------------
__global__ void k_init_agg(unsigned int* agg, long n) {
  long i = (long)blockIdx.x * blockDim.x + threadIdx.x;
  long stride = (long)gridDim.x * blockDim.x;
  for (; i < n; i += stride) agg[i] = 0xFF800000u;  // -inf bits
}

// ---------------------------------------------------------------------------
// Kernel 1: per-edge MLP via WMMA + atomic max scatter to agg[N, 32]
// One wave handles a tile of 16 edges.
// ---------------------------------------------------------------------------
__global__ void __launch_bounds__(256)
k_edge_mlp(const float* __restrict__ x,
           const long long* __restrict__ ei,   // [2, E] int64 (src row, dst row)
           const float* __restrict__ W1, const float* __restrict__ b1,
           const float* __restrict__ W2, const float* __restrict__ b2,
           float* __restrict__ agg, int E) {
  __shared__ float sH[WAVES_PER_BLOCK][16 * LDS_STRIDE];
  __shared__ int   sDst[WAVES_PER_BLOCK][16];

  const int lane = threadIdx.x & 31;
  const int wave = threadIdx.x >> 5;
  const int m    = lane & 15;       // edge-in-tile / column index
  const int hi   = lane >> 4;       // upper half of wave -> K offset +2 / rows 8..15

  // ---- Load weight B-fragments (held in registers for all tiles) ----------
  // B layout for 16x16x4 f32: lane = N column; v0 = B[k0][n], v1 = B[k0+1][n];
  // lanes 16..31 use K offsets +2/+3 of the chunk.
  v2f w1f[2][2];                    // [k-chunk][col-block], W1 is 8x32
  for (int kc = 0; kc < 2; ++kc)
    for (int nb = 0; nb < 2; ++nb) {
      int k0 = kc * 4 + hi * 2;
      int n  = nb * 16 + m;
      w1f[kc][nb] = (v2f){ W1[k0 * HID + n], W1[(k0 + 1) * HID + n] };
    }
  v2f w2f[8][2];                    // W2 is 32x32
  for (int kc = 0; kc < 8; ++kc)
    for (int nb = 0; nb < 2; ++nb) {
      int k0 = kc * 4 + hi * 2;
      int n  = nb * 16 + m;
      w2f[kc][nb] = (v2f){ W2[k0 * HID + n], W2[(k0 + 1) * HID + n] };
    }
  const float bias1_0 = b1[m], bias1_1 = b1[16 + m];
  const float bias2_0 = b2[m], bias2_1 = b2[16 + m];

  const int ntiles = (E + 15) >> 4;
  const int wstride = gridDim.x * WAVES_PER_BLOCK;
  float* const hrow = sH[wave];

  for (int t = blockIdx.x * WAVES_PER_BLOCK + wave; t < ntiles; t += wstride) {
    // ---- Gather one tile of 16 edges (both half-waves load same edges) ----
    int e  = t * 16 + m;
    int ec = e < E ? e : (E - 1);
    int src = (int)ei[ec];
    int dst = (int)ei[(long)E + ec];
    if (lane < 16) sDst[wave][m] = dst;

    const float4 xi = ((const float4*)x)[dst];
    const float4 xj = ((const float4*)x)[src];

    // A-fragments: msg = [xi, xj - xi]; chunk0 = msg[0..3], chunk1 = msg[4..7]
    v2f a0, a1;
    if (hi == 0) {
      a0 = (v2f){ xi.x, xi.y };
      a1 = (v2f){ xj.x - xi.x, xj.y - xi.y };
    } else {
      a0 = (v2f){ xi.z, xi.w };
      a1 = (v2f){ xj.z - xi.z, xj.w - xi.w };
    }

    // ---- Layer 1: H1 = relu(msg @ W1 + b1)  [16 x 32] ---------------------
    v8f c0, c1;
    for (int r = 0; r < 8; ++r) { c0[r] = bias1_0; c1[r] = bias1_1; }
    c0 = wmma_f32_16x16x4(a0, w1f[0][0], c0);
    c0 = wmma_f32_16x16x4(a1, w1f[1][0], c0);
    c1 = wmma_f32_16x16x4(a0, w1f[0][1], c1);
    c1 = wmma_f32_16x16x4(a1, w1f[1][1], c1);
    for (int r = 0; r < 8; ++r) {
      c0[r] = fmaxf(c0[r], 0.0f);
      c1[r] = fmaxf(c1[r], 0.0f);
    }

    // ---- Transpose C layout -> A layout through per-wave LDS tile ---------
    for (int r = 0; r < 8; ++r) {
      int row = hi * 8 + r;                       // C: lanes 16..31 hold M=8..15
      hrow[row * LDS_STRIDE + m]      = c0[r];    // cols 0..15
      hrow[row * LDS_STRIDE + 16 + m] = c1[r];    // cols 16..31
    }
    v2f a2[8];
    for (int kc = 0; kc < 8; ++kc) {
      int kb = kc * 4 + hi * 2;
      a2[kc] = (v2f){ hrow[m * LDS_STRIDE + kb], hrow[m * LDS_STRIDE + kb + 1] };
    }

    // ---- Layer 2: H2 = H1 @ W2 + b2  [16 x 32] ----------------------------
    v8f d0, d1;
    for (int r = 0; r < 8; ++r) { d0[r] = bias2_0; d1[r] = bias2_1; }
    for (int kc = 0; kc < 8; ++kc) d0 = wmma_f32_16x16x4(a2[kc], w2f[kc][0], d0);
    for (int kc = 0; kc < 8; ++kc) d1 = wmma_f32_16x16x4(a2[kc], w2f[kc][1], d1);

    // ---- Native atomic segment-max into agg[dst, :] -----------------------
    for (int r = 0; r < 8; ++r) {
      int row  = hi * 8 + r;
      int erow = t * 16 + row;
      if (erow < E) {
        int drow = sDst[wave][row];
        atomicMaxF(&agg[(long)drow * HID + m],      d0[r]);
        atomicMaxF(&agg[(long)drow * HID + 16 + m], d1[r]);
      }
    }
  }
}

// ---------------------------------------------------------------------------
// Kernel 2: replace non-finite agg with 0; deterministic partial BN stats.
// thread t: channel c = t&31, node-lane r = t>>5; block tree reduce over r.
// ---------------------------------------------------------------------------
__global__ void __launch_bounds__(256)
k_bn_partial(float* __restrict__ agg, float* __restrict__ partial, int N) {
  const int c = threadIdx.x & 31;
  const int r = threadIdx.x >> 5;
  float s = 0.0f, q = 0.0f;
  for (long n = (long)blockIdx.x * 8 + r; n < N; n += (long)gridDim.x * 8) {
    float v = agg[n * HID + c];
    if (!isfinite(v)) { v = 0.0f; agg[n * HID + c] = 0.0f; }
    s += v;
    q += v * v;
  }
  __shared__ float ss[8][HID], sq[8][HID];
  ss[r][c] = s; sq[r][c] = q;
  __syncthreads();
  if (r == 0) {
    float S = 0.0f, Q = 0.0f;
    for (int k = 0; k < 8; ++k) { S += ss[k][c]; Q += sq[k][c]; }  // fixed order
    partial[blockIdx.x * 64 + c]      = S;
    partial[blockIdx.x * 64 + 32 + c] = Q;
  }
}

// ---------------------------------------------------------------------------
// Kernel 3: finalize BN -> scale/shift per channel (deterministic order)
// ---------------------------------------------------------------------------
__global__ void k_bn_finalize(const float* __restrict__ partial,
                              const float* __restrict__ gamma,
                              const float* __restrict__ beta,
                              float* __restrict__ scale,
                              float* __restrict__ shift, int N, int nblk) {
  int c = threadIdx.x;
  if (c < HID) {
    float S = 0.0f, Q = 0.0f;
    for (int b = 0; b < nblk; ++b) {
      S += partial[b * 64 + c];
      Q += partial[b * 64 + 32 + c];
    }
    float invN = 1.0f / (float)N;
    float mean = S * invN;
    float var  = fmaxf(Q * invN - mean * mean, 0.0f);
    float sc   = gamma[c] * rsqrtf(var + 1e-5f);
    scale[c] = sc;
    shift[c] = beta[c] - mean * sc;
  }
}

// ---------------------------------------------------------------------------
// Graph boundary detection (batch is sorted): start[g] = first node of graph g
// ---------------------------------------------------------------------------
__global__ void k_bounds_init(int* __restrict__ start, int G, int N) {
  int i = threadIdx.x;
  if (i <= G) start[i] = N;
}

__global__ void k_bounds_scan(const long long* __restrict__ batch,
                              int* __restrict__ start, int N) {
  long i = (long)blockIdx.x * blockDim.x + threadIdx.x;
  long stride = (long)gridDim.x * blockDim.x;
  for (; i < N; i += stride) {
    int g = (int)batch[i];
    if (i == 0 || (int)batch[i - 1] != g) start[g] = (int)i;  // unique writer
  }
}

__global__ void k_bounds_fix(int* __restrict__ start, int G, int N) {
  if (threadIdx.x == 0) {
    start[G] = N;
    for (int g = G - 1; g >= 0; --g)          // make monotone; empty -> empty range
      if (start[g] > start[g + 1]) start[g] = start[g + 1];
  }
}

// ---------------------------------------------------------------------------
// Kernel 4: per-graph pooling over [start[g], start[g+1]) — single pass over
// agg in total, deterministic fixed-order tree reduction.
// ---------------------------------------------------------------------------
__global__ void __launch_bounds__(256)
k_pool(const float* __restrict__ agg, const int* __restrict__ start,
       const float* __restrict__ scale, const float* __restrict__ shift,
       float* __restrict__ pooled, float* __restrict__ counts) {
  const int g = blockIdx.x;
  const int c = threadIdx.x & 31;
  const int r = threadIdx.x >> 5;
  const int lo = start[g], hi = start[g + 1];
  const float sc = scale[c], sh = shift[c];
  float s = 0.0f;
  for (long n = lo + r; n < hi; n += 8) {
    float v = agg[n * HID + c] * sc + sh;
    s += fmaxf(v, 0.0f);
  }
  __shared__ float ss[8][HID];
  ss[r][c] = s;
  __syncthreads();
  if (r == 0) {
    float S = 0.0f;
    for (int k = 0; k < 8; ++k) S += ss[k][c];   // fixed order
    pooled[g * HID + c] = S;
    if (c == 0) counts[g] = (float)(hi - lo);
  }
}

// ---------------------------------------------------------------------------
// Kernel 5: head — relu(pooled/cnt @ Wr1 + br1) @ Wr2 + br2  -> out[G]
// ---------------------------------------------------------------------------
__global__ void k_head(const float* __restrict__ pooled,
                       const float* __restrict__ counts,
                       const float* __restrict__ Wr1, const float* __restrict__ br1,
                       const float* __restrict__ Wr2, const float* __restrict__ br2,
                       float* __restrict__ out, int G) {
  int g = threadIdx.x;
  if (g < G) {
    float inv = 1.0f / fmaxf(counts[g], 1.0f);
    float p[HID];
    for (int c = 0; c < HID; ++c) p[c] = pooled[g * HID + c] * inv;
    float o = br2[0];
    for (int j = 0; j < 16; ++j) {
      float h = br1[j];
      for (int c = 0; c < HID; ++c) h += p[c] * Wr1[c * 16 + j];
      o += fmaxf(h, 0.0f) * Wr2[j];
    }
    out[g] = o;
  }
}

// ---------------------------------------------------------------------------
extern "C" void kernel_launch(void* const* d_in, const int* in_sizes, int n_in,
                              void* d_out, int out_size, void* d_ws, size_t ws_size,
                              hipStream_t stream) {
  const float*     x     = (const float*)d_in[0];
  const long long* ei    = (const long long*)d_in[1];
  const long long* batch = (const long long*)d_in[2];
  const float* W1    = (const float*)d_in[3];
  const float* b1    = (const float*)d_in[4];
  const float* W2    = (const float*)d_in[5];
  const float* b2    = (const float*)d_in[6];
  const float* gamma = (const float*)d_in[7];
  const float* beta  = (const float*)d_in[8];
  const float* Wr1   = (const float*)d_in[9];
  const float* br1   = (const float*)d_in[10];
  const float* Wr2   = (const float*)d_in[11];
  const float* br2   = (const float*)d_in[12];
  float* out = (float*)d_out;

  const int N = in_sizes[0] / 4;   // IN_DIM = 4
  const int E = in_sizes[1] / 2;   // edge_index is [2, E]
  const int G = out_size;          // 64 graphs

  // Workspace layout
  float* agg     = (float*)d_ws;                 // N * 32 floats
  float* partial = agg + (long)N * HID;          // STAT_BLOCKS * 64
  float* scale   = partial + STAT_BLOCKS * 64;   // 32
  float* shift   = scale + HID;                  // 32
  float* pooled  = shift + HID;                  // G * 32
  float* counts  = pooled + (long)G * HID;       // G
  int*   start   = (int*)(counts + G);           // G + 1 ints

  k_init_agg<<<1024, 256, 0, stream>>>((unsigned int*)agg, (long)N * HID);
  k_bounds_init<<<1, 128, 0, stream>>>(start, G, N);
  k_bounds_scan<<<256, 256, 0, stream>>>(batch, start, N);
  k_bounds_fix<<<1, 32, 0, stream>>>(start, G, N);
  k_edge_mlp<<<EDGE_BLOCKS, 256, 0, stream>>>(x, ei, W1, b1, W2, b2, agg, E);
  k_bn_partial<<<STAT_BLOCKS, 256, 0, stream>>>(agg, partial, N);
  k_bn_finalize<<<1, 32, 0, stream>>>(partial, gamma, beta, scale, shift, N, STAT_BLOCKS);
  k_pool<<<G, 256, 0, stream>>>(agg, start, scale, shift, pooled, counts);
  k_head<<<1, 64, 0, stream>>>(pooled, counts, Wr1, br1, Wr2, br2, out, G);
}